// GCN_63866163691820
// MI455X (gfx1250) — compile-verified
//
#include <hip/hip_runtime.h>
#include <hip/hip_bf16.h>

typedef float v2f __attribute__((ext_vector_type(2)));
typedef float v8f __attribute__((ext_vector_type(8)));

#define N_NODES 100000
#define N_EDGES 1600000
#define IN_F 64
#define HID 64
#define NC 40

// ---------------------------------------------------------------------------
// Zero-fill (agg buffer must start at 0 before atomic accumulation)
// ---------------------------------------------------------------------------
__global__ void gcn_zero_kernel(float* __restrict__ p, int n) {
  int i = blockIdx.x * blockDim.x + threadIdx.x;
  if (i < n) p[i] = 0.0f;
}

// ---------------------------------------------------------------------------
// Seed output with bias b2 (out = A*(h@W2) + b2; isolated nodes keep b2)
// ---------------------------------------------------------------------------
__global__ void gcn_bias_init_kernel(float* __restrict__ out,
                                     const float* __restrict__ b) {
  unsigned i = blockIdx.x * blockDim.x + threadIdx.x;
  const unsigned total = (unsigned)N_NODES * NC;
  if (i < total) {
    unsigned c = i - (i / NC) * NC;
    out[i] = b[c];
  }
}

// ---------------------------------------------------------------------------
// Edge scatter-add: acc[dst[e], :] += x[src[e], :]
// One thread per (edge, 4-float group): coalesced float4 gather (L2-resident)
// + four global_atomic_add_f32 into L2.
// ---------------------------------------------------------------------------
__global__ void gcn_scatter_add_kernel(const float* __restrict__ x,
                                       const int* __restrict__ src,
                                       const int* __restrict__ dst,
                                       float* __restrict__ acc,
                                       int groups, int stride) {
  unsigned tid = blockIdx.x * blockDim.x + threadIdx.x;
  unsigned total = (unsigned)N_EDGES * (unsigned)groups;
  if (tid >= total) return;
  unsigned e = tid / (unsigned)groups;
  unsigned g = tid - e * (unsigned)groups;
  int s = src[e];
  int d = dst[e];
  const float4 v =
      *reinterpret_cast<const float4*>(x + (size_t)s * stride + (size_t)g * 4);
  float* o = acc + (size_t)d * stride + (size_t)g * 4;
  atomicAdd(o + 0, v.x);
  atomicAdd(o + 1, v.y);
  atomicAdd(o + 2, v.z);
  atomicAdd(o + 3, v.w);
}

// ---------------------------------------------------------------------------
// Dense GEMM via V_WMMA_F32_16X16X4_F32: C[M x NREAL] = A[M x KDIM] @ W + bias
// One wave per 16x16 output tile; K loop in steps of 4, fully unrolled.
// A frag (16x4 f32): lanes 0-15 = M rows, VGPR0/1 = K {0,1} (low half),
//                    {2,3} (high half).  B frag mirrors with lane = N column.
// FULL = (NREAL % 16 == 0): compile-time full tiles, no column masking at all
// (removes the mask-multiply from the WMMA dependency chain in layer 1).
// Otherwise mask via clamped index + multiply so EXEC stays all-1s around the
// WMMA ops (ISA requirement).
// ---------------------------------------------------------------------------
template <int KDIM, int NREAL, bool RELU, bool BIAS>
__global__ void gcn_wmma_gemm_kernel(const float* __restrict__ A,
                                     const float* __restrict__ W,
                                     const float* __restrict__ bias,
                                     float* __restrict__ C,
                                     int n_row_tiles, int n_col_tiles) {
  constexpr bool FULL = (NREAL % 16 == 0);

  const int tile = blockIdx.x * (blockDim.x >> 5) + (threadIdx.x >> 5);
  const int n_tiles = n_row_tiles * n_col_tiles;
  if (tile >= n_tiles) return;  // whole-wave exit: EXEC all-1s for WMMA

  const int row_tile = tile / n_col_tiles;
  const int col_tile = tile - row_tile * n_col_tiles;
  const int row0 = row_tile << 4;
  const int col0 = col_tile << 4;

  const int lane = threadIdx.x & 31;
  const int lh = lane & 15;  // M index (A) / N index (B)
  const int hi = lane >> 4;  // selects K pair {0,1} vs {2,3}

  const int nb = col0 + lh;
  const bool nok = FULL ? true : (nb < NREAL);
  const int nbc = FULL ? nb : (nok ? nb : (NREAL - 1));
  const float nmask = nok ? 1.0f : 0.0f;

  const float* arow = A + (size_t)(row0 + lh) * KDIM + (size_t)(hi * 2);

  v8f c = {};
#pragma unroll
  for (int k = 0; k < KDIM; k += 4) {
    v2f a, b;
    a.x = arow[k];
    a.y = arow[k + 1];
    const int kb = k + hi * 2;
    if (FULL) {
      b.x = W[(size_t)kb * NREAL + nbc];
      b.y = W[(size_t)(kb + 1) * NREAL + nbc];
    } else {
      b.x = W[(size_t)kb * NREAL + nbc] * nmask;
      b.y = W[(size_t)(kb + 1) * NREAL + nbc] * nmask;
    }
    // D = A*B + C  (neg_a, A, neg_b, B, c_mod, C, reuse_a, reuse_b)
    c = __builtin_amdgcn_wmma_f32_16x16x4_f32(false, a, false, b, (short)0, c,
                                              false, false);
  }

  if (!nok) return;  // past all WMMA ops; per-lane exit is safe now
  float bv = 0.0f;
  if (BIAS) bv = bias[nb];
#pragma unroll
  for (int r = 0; r < 8; ++r) {
    const int row = row0 + r + (hi << 3);  // C layout: VGPR r -> M = r (+8 hi)
    float v = c[r] + bv;
    if (RELU) v = fmaxf(v, 0.0f);
    C[(size_t)row * NREAL + nb] = v;
  }
}

// ---------------------------------------------------------------------------
// Launch:  inputs: features, src, dst, W1, b1, W2, b2
//   agg1 = ws[0 .. 6.4M)          (25.6 MB)
//   h    = ws[6.4M .. 12.8M)      (25.6 MB)
//   t    = ws[0 .. 4.0M)          aliases agg1 (dead after GEMM-1)
// ---------------------------------------------------------------------------
extern "C" void kernel_launch(void* const* d_in, const int* in_sizes, int n_in,
                              void* d_out, int out_size, void* d_ws,
                              size_t ws_size, hipStream_t stream) {
  (void)in_sizes; (void)n_in; (void)out_size; (void)ws_size;
  const float* features = (const float*)d_in[0];
  const int*   src      = (const int*)d_in[1];
  const int*   dst      = (const int*)d_in[2];
  const float* W1       = (const float*)d_in[3];
  const float* b1       = (const float*)d_in[4];
  const float* W2       = (const float*)d_in[5];
  const float* b2       = (const float*)d_in[6];
  float* out = (float*)d_out;

  float* ws   = (float*)d_ws;
  float* agg1 = ws;                          // N*64 floats
  float* h    = ws + (size_t)N_NODES * HID;  // N*64 floats
  float* t    = ws;                          // N*40 floats, reuses agg1

  const int nAgg1 = N_NODES * IN_F;  // 6,400,000

  // 1) agg1 = 0
  gcn_zero_kernel<<<(nAgg1 + 255) / 256, 256, 0, stream>>>(agg1, nAgg1);

  // 2) agg1[dst] += features[src]   (16 float4-groups per edge)
  {
    const int total = N_EDGES * 16;
    gcn_scatter_add_kernel<<<(total + 255) / 256, 256, 0, stream>>>(
        features, src, dst, agg1, 16, IN_F);
  }

  // 3) h = relu(agg1 @ W1 + b1)   -- WMMA f32 16x16x4, 6250x4 tiles, unmasked
  {
    const int rt = N_NODES / 16, ct = HID / 16;  // 6250, 4
    const int tiles = rt * ct;
    gcn_wmma_gemm_kernel<HID, HID, true, true>
        <<<(tiles + 7) / 8, 256, 0, stream>>>(agg1, W1, b1, h, rt, ct);
  }

  // 4) t = h @ W2  (bias deferred to output seed) -- 6250x3 tiles, 40 cols
  {
    const int rt = N_NODES / 16, ct = (NC + 15) / 16;  // 6250, 3
    const int tiles = rt * ct;
    gcn_wmma_gemm_kernel<HID, NC, false, false>
        <<<(tiles + 7) / 8, 256, 0, stream>>>(h, W2, nullptr, t, rt, ct);
  }

  // 5) out[i, :] = b2
  {
    const int total = N_NODES * NC;
    gcn_bias_init_kernel<<<(total + 255) / 256, 256, 0, stream>>>(out, b2);
  }

  // 6) out[dst] += t[src]   (10 float4-groups per edge: 40 atomics/edge)
  {
    const int total = N_EDGES * 10;
    gcn_scatter_add_kernel<<<(total + 255) / 256, 256, 0, stream>>>(
        t, src, dst, out, 10, NC);
  }
}